// Graph_GSN_6571299963189
// MI455X (gfx1250) — compile-verified
//
#include <hip/hip_runtime.h>
#include <hip/hip_bf16.h>

#define N_NODES   100000
#define N_EDGES   1000000
#define NUM_GRAPHS 512
#define HIDDEN    64
#define ID_DIM    16
#define IN0       128            // layer 0 input dim
#define INL       80             // ID_DIM + HIDDEN for layers 1..3
#define NBLK      782            // ceil(N_NODES / 128)
#define NPAD      (NBLK * 128)   // 100096 padded node rows
#define OUT_COLS  256            // HIDDEN * NUM_LAYERS
#define BN_EPS    1e-5f

typedef __attribute__((ext_vector_type(2))) float v2f;
typedef __attribute__((ext_vector_type(8))) float v8f;

// Hardware fp32 atomic add (global_atomic_add_f32); device memory is not
// fine-grained here, and the reference has no ordering requirement on the sums.
__device__ __forceinline__ void atomAddF(float* p, float v) {
    (void)unsafeAtomicAdd(p, v);
}

// ---------------------------------------------------------------- utilities
__global__ void zero_f32(float* __restrict__ p, int n) {
    int i = blockIdx.x * blockDim.x + threadIdx.x;
    if (i < n) p[i] = 0.f;
}

// Layer 0: agg[row][0..127] = (row < N) ? x[row] : 0   (padded copy; h = x + sum)
__global__ void init_agg0(const float* __restrict__ x, float* __restrict__ agg) {
    int i = blockIdx.x * blockDim.x + threadIdx.x;   // NPAD * 32 float4 chunks
    int row = i >> 5, ch = i & 31;
    if (row >= NPAD) return;
    float4 v = make_float4(0.f, 0.f, 0.f, 0.f);
    if (row < N_NODES) v = ((const float4*)x)[row * 32 + ch];
    ((float4*)agg)[row * 32 + ch] = v;
}

// Layers 1..3: z_in[row] = concat(x[row,:16], z_prev[row]) ; agg = copy of z_in
__global__ void concat_init(const float* __restrict__ x,
                            const float* __restrict__ zout,  // d_out z region, stride 256
                            int prev_col,
                            float* __restrict__ zin,         // [NPAD, 80]
                            float* __restrict__ agg) {       // [NPAD, 80]
    int i = blockIdx.x * blockDim.x + threadIdx.x;   // NPAD * 20 float4 chunks
    int row = i / 20, ch = i % 20;
    if (row >= NPAD) return;
    float4 v = make_float4(0.f, 0.f, 0.f, 0.f);
    if (row < N_NODES) {
        if (ch < 4) v = ((const float4*)x)[row * 32 + ch];
        else        v = *(const float4*)(zout + (long long)row * OUT_COLS + prev_col + (ch - 4) * 4);
    }
    ((float4*)zin)[row * 20 + ch] = v;
    ((float4*)agg)[row * 20 + ch] = v;
}

// agg[dst] += z_in[src] over all edges; thread = (edge, float4 chunk)
__global__ void edge_scatter(const float* __restrict__ zin,
                             const int* __restrict__ ei,   // [2, E]
                             float* __restrict__ agg,
                             int chunks, int in_dim) {
    int i = blockIdx.x * blockDim.x + threadIdx.x;
    if (i >= N_EDGES * chunks) return;
    int e = i / chunks, ch = i % chunks;
    int src = ei[e];
    int dst = ei[N_EDGES + e];
    float4 v = *(const float4*)(zin + (long long)src * in_dim + ch * 4);
    float* d = agg + (long long)dst * in_dim + ch * 4;
    atomAddF(d + 0, v.x);
    atomAddF(d + 1, v.y);
    atomAddF(d + 2, v.z);
    atomAddF(d + 3, v.w);
}

// ---------------------------------------------------------------- GEMM 1 (WMMA f32)
// h1 = relu(h @ W1 + b1); block = 256 threads = 8 waves; wave -> 16 rows x 64 cols.
// W1 is staged in LDS pre-paired along K so each lane's B-fragment (b.x=W[k'][n],
// b.y=W[k'+1][n], k' even) is a single aligned ds_load_b64.
__global__ __launch_bounds__(256) void mlp1_wmma(
    const float* __restrict__ h,    // [NPAD, in_dim]
    int in_dim,
    const float* __restrict__ W1,   // [in_dim, 64]
    const float* __restrict__ b1,   // [64]
    float* __restrict__ h1out) {    // [NPAD, 64]
    __shared__ v2f W1p[(IN0 / 2) * HIDDEN];   // up to 32 KB; W1p[kp*64+n] = (W1[2kp][n], W1[2kp+1][n])

    int tid = threadIdx.x;
    for (int i = tid; i < (in_dim / 2) * HIDDEN; i += 256) {
        int kp = i >> 6, n = i & 63;
        v2f w;
        w.x = W1[(2 * kp) * HIDDEN + n];
        w.y = W1[(2 * kp + 1) * HIDDEN + n];
        W1p[i] = w;
    }
    __syncthreads();

    int lane = tid & 31, wave = tid >> 5;
    int lrow = lane & 15;        // row within A tile / col within B,D tile
    int kh   = lane >> 4;        // K-half select (ISA 16x16x4 f32 layouts)
    long long rowBase = (long long)blockIdx.x * 128 + wave * 16;

    const float* arow = h + (rowBase + lrow) * in_dim + 2 * kh;
    const v2f*   brow = W1p + kh * HIDDEN + lrow;   // + (k/2)*64 + tile*16 per step
    v8f a0 = {}, a1 = {}, a2 = {}, a3 = {};
    for (int k = 0; k < in_dim; k += 4) {
        v2f av = *(const v2f*)(arow + k);
        const v2f* bk = brow + (k >> 1) * HIDDEN;
        a0 = __builtin_amdgcn_wmma_f32_16x16x4_f32(false, av, false, bk[0],  (short)0, a0, false, false);
        a1 = __builtin_amdgcn_wmma_f32_16x16x4_f32(false, av, false, bk[16], (short)0, a1, false, false);
        a2 = __builtin_amdgcn_wmma_f32_16x16x4_f32(false, av, false, bk[32], (short)0, a2, false, false);
        a3 = __builtin_amdgcn_wmma_f32_16x16x4_f32(false, av, false, bk[48], (short)0, a3, false, false);
    }

    auto emit = [&](v8f c, int t) {
        int col = t * 16 + lrow;
        float bb = b1[col];
#pragma unroll
        for (int v = 0; v < 8; ++v) {
            long long r = rowBase + v + 8 * kh;          // D layout: row = v + 8*(lane>>4)
            float val = c[v] + bb;
            h1out[r * HIDDEN + col] = val > 0.f ? val : 0.f;
        }
    };
    emit(a0, 0); emit(a1, 1); emit(a2, 2); emit(a3, 3);
}

// ---------------------------------------------------------------- GEMM 2 (WMMA f32) + BN stats
// h2 = h1 @ W2 + b2; also accumulate per-channel sum / sumsq over valid rows
__global__ __launch_bounds__(256) void mlp2_wmma(
    const float* __restrict__ h1,   // [NPAD, 64]
    const float* __restrict__ W2,   // [64, 64]
    const float* __restrict__ b2,   // [64]
    float* __restrict__ h2out,      // [NPAD, 64]
    float* __restrict__ stats) {    // [128]: sum[64], sumsq[64]
    __shared__ v2f   W2p[(HIDDEN / 2) * HIDDEN];  // 16 KB, K-paired as in mlp1
    __shared__ float h2s[128 * HIDDEN];           // 32 KB, block's h2 tile for reduction

    int tid = threadIdx.x;
    for (int i = tid; i < (HIDDEN / 2) * HIDDEN; i += 256) {
        int kp = i >> 6, n = i & 63;
        v2f w;
        w.x = W2[(2 * kp) * HIDDEN + n];
        w.y = W2[(2 * kp + 1) * HIDDEN + n];
        W2p[i] = w;
    }
    __syncthreads();

    int lane = tid & 31, wave = tid >> 5;
    int lrow = lane & 15, kh = lane >> 4;
    long long rowBase = (long long)blockIdx.x * 128 + wave * 16;

    const float* arow = h1 + (rowBase + lrow) * HIDDEN + 2 * kh;
    const v2f*   brow = W2p + kh * HIDDEN + lrow;
    v8f a0 = {}, a1 = {}, a2 = {}, a3 = {};
#pragma unroll
    for (int k = 0; k < HIDDEN; k += 4) {
        v2f av = *(const v2f*)(arow + k);
        const v2f* bk = brow + (k >> 1) * HIDDEN;
        a0 = __builtin_amdgcn_wmma_f32_16x16x4_f32(false, av, false, bk[0],  (short)0, a0, false, false);
        a1 = __builtin_amdgcn_wmma_f32_16x16x4_f32(false, av, false, bk[16], (short)0, a1, false, false);
        a2 = __builtin_amdgcn_wmma_f32_16x16x4_f32(false, av, false, bk[32], (short)0, a2, false, false);
        a3 = __builtin_amdgcn_wmma_f32_16x16x4_f32(false, av, false, bk[48], (short)0, a3, false, false);
    }

    auto emit = [&](v8f c, int t) {
        int col = t * 16 + lrow;
        float bb = b2[col];
#pragma unroll
        for (int v = 0; v < 8; ++v) {
            int rl = wave * 16 + v + 8 * kh;
            float val = c[v] + bb;
            h2s[rl * HIDDEN + col] = val;
            h2out[(rowBase + v + 8 * kh) * HIDDEN + col] = val;
        }
    };
    emit(a0, 0); emit(a1, 1); emit(a2, 2); emit(a3, 3);
    __syncthreads();

    // per-channel partial reduction over this block's 128 rows (mask padded rows)
    int c  = tid & 63;
    int rq = tid >> 6;                    // 4 threads per channel
    float s = 0.f, s2 = 0.f;
    for (int r = rq * 32; r < rq * 32 + 32; ++r) {
        long long grow = (long long)blockIdx.x * 128 + r;
        if (grow < N_NODES) {
            float v = h2s[r * HIDDEN + c];
            s += v; s2 += v * v;
        }
    }
    atomAddF(&stats[c], s);
    atomAddF(&stats[64 + c], s2);
}

// ---------------------------------------------------------------- BN finalize + apply
__global__ void bn_finalize(const float* __restrict__ stats,
                            const float* __restrict__ gamma,
                            const float* __restrict__ beta,
                            float* __restrict__ ss) {   // [128]: scale, shift
    int c = threadIdx.x;
    if (c >= HIDDEN) return;
    const float inv_n = 1.f / (float)N_NODES;
    float mu  = stats[c] * inv_n;
    float var = stats[64 + c] * inv_n - mu * mu;
    float rs  = rsqrtf(var + BN_EPS);
    float sc  = gamma[c] * rs;
    ss[c]      = sc;
    ss[64 + c] = beta[c] - mu * sc;
}

// z = relu(h2*scale + shift) -> z_out columns; atomic pool into g_out by graph id
__global__ void bn_apply_pool(const float* __restrict__ h2,
                              const float* __restrict__ ss,
                              const int* __restrict__ batch,
                              float* __restrict__ zout,   // [N, 256]
                              float* __restrict__ gout,   // [512, 256]
                              int col0) {
    int i = blockIdx.x * blockDim.x + threadIdx.x;   // N_NODES * 16 float4 chunks
    if (i >= N_NODES * 16) return;
    int node = i >> 4, c0 = (i & 15) * 4;
    float4 v  = *(const float4*)(h2 + (long long)node * HIDDEN + c0);
    float4 sc = *(const float4*)(ss + c0);
    float4 sh = *(const float4*)(ss + 64 + c0);
    float4 z;
    z.x = fmaxf(v.x * sc.x + sh.x, 0.f);
    z.y = fmaxf(v.y * sc.y + sh.y, 0.f);
    z.z = fmaxf(v.z * sc.z + sh.z, 0.f);
    z.w = fmaxf(v.w * sc.w + sh.w, 0.f);
    *(float4*)(zout + (long long)node * OUT_COLS + col0 + c0) = z;
    float* g = gout + (long long)batch[node] * OUT_COLS + col0 + c0;
    atomAddF(g + 0, z.x);
    atomAddF(g + 1, z.y);
    atomAddF(g + 2, z.z);
    atomAddF(g + 3, z.w);
}

// ---------------------------------------------------------------- launch
static inline int cdiv(long long a, int b) { return (int)((a + b - 1) / b); }

extern "C" void kernel_launch(void* const* d_in, const int* in_sizes, int n_in,
                              void* d_out, int out_size, void* d_ws, size_t ws_size,
                              hipStream_t stream) {
    (void)in_sizes; (void)n_in; (void)out_size; (void)ws_size;

    const float* x     = (const float*)d_in[0];
    const int*   ei    = (const int*)d_in[1];
    const int*   batch = (const int*)d_in[2];

    float* zout = (float*)d_out;                              // [N, 256]
    float* gout = zout + (size_t)N_NODES * OUT_COLS;          // [512, 256]

    // workspace carve-up (aliased buffers, lifetimes are stream-ordered):
    //   agg : [NPAD,128]  (h = z + sum)        -> reused as h2 buffer [NPAD,64]
    //   zin : [NPAD, 80]  (concat input)       -> reused as h1 buffer [NPAD,64]
    float* agg   = (float*)d_ws;
    float* zin   = agg + (size_t)NPAD * IN0;
    float* h1buf = zin;
    float* h2buf = agg;
    float* stats = zin + (size_t)NPAD * INL;                  // [128]
    float* ss    = stats + 128;                                // [128]

    zero_f32<<<cdiv(NUM_GRAPHS * OUT_COLS, 256), 256, 0, stream>>>(gout, NUM_GRAPHS * OUT_COLS);

    for (int l = 0; l < 4; ++l) {
        const int in_dim = (l == 0) ? IN0 : INL;
        const float* W1    = (const float*)d_in[3 + 6 * l + 0];
        const float* b1    = (const float*)d_in[3 + 6 * l + 1];
        const float* W2    = (const float*)d_in[3 + 6 * l + 2];
        const float* b2    = (const float*)d_in[3 + 6 * l + 3];
        const float* gamma = (const float*)d_in[3 + 6 * l + 4];
        const float* beta  = (const float*)d_in[3 + 6 * l + 5];

        const float* zsrc;
        if (l == 0) {
            init_agg0<<<cdiv((long long)NPAD * 32, 256), 256, 0, stream>>>(x, agg);
            zsrc = x;
        } else {
            concat_init<<<cdiv((long long)NPAD * 20, 256), 256, 0, stream>>>(
                x, zout, (l - 1) * HIDDEN, zin, agg);
            zsrc = zin;
        }

        const int chunks = in_dim / 4;
        edge_scatter<<<cdiv((long long)N_EDGES * chunks, 256), 256, 0, stream>>>(
            zsrc, ei, agg, chunks, in_dim);

        mlp1_wmma<<<NBLK, 256, 0, stream>>>(agg, in_dim, W1, b1, h1buf);

        zero_f32<<<1, 128, 0, stream>>>(stats, 128);
        mlp2_wmma<<<NBLK, 256, 0, stream>>>(h1buf, W2, b2, h2buf, stats);

        bn_finalize<<<1, 64, 0, stream>>>(stats, gamma, beta, ss);
        bn_apply_pool<<<cdiv((long long)N_NODES * 16, 256), 256, 0, stream>>>(
            h2buf, ss, batch, zout, gout, l * HIDDEN);
    }
}